// Greedy_64029372449327
// MI455X (gfx1250) — compile-verified
//
#include <hip/hip_runtime.h>

// Problem: L=2048 symmetrize+mask, then greedy max-matching (<=1024 pairs),
// output keeps only picked (i,j)/(j,i) entries. Memory-bound (~80MB traffic
// total -> ~3.5us at 23.3 TB/s HBM); greedy is tiny sequential work done by
// one 1024-thread workgroup with lazy per-row maxima.
//
// CDNA5 specifics: V_WMMA_F32_16X16X4_F32 with identity operands performs the
// cross-lane 16x16 f32 transpose shuffle (A-layout -> D-layout) so the
// symmetrize kernel needs no LDS staging; both the straight tile and the
// transposed-block tile are read with coalesced 64B half-row accesses.

#define MIN_DIST 4

typedef float v2f __attribute__((ext_vector_type(2)));
typedef float v8f __attribute__((ext_vector_type(8)));

// ---- K1: per-position base codes from argmax over 4 channels --------------
__global__ void pairs_kernel(const float* __restrict__ feat,
                             int* __restrict__ codes, int L) {
  int i = blockIdx.x * blockDim.x + threadIdx.x;
  if (i >= L) return;
  const int L2 = L * L;
  float best = feat[i * L];            // ch 0, col 0
  int idx = 0;
#pragma unroll
  for (int ch = 1; ch < 4; ++ch) {
    float v = feat[ch * L2 + i * L];
    if (v > best) { best = v; idx = ch; }  // first-max tie rule (strict >)
  }
  const int bases[4] = {2, 3, 5, 7};
  codes[i] = bases[idx];
}

// ---- K2: symmetrize + band + pair masks via WMMA transpose ----------------
// One wave per 16x16 tile; 8 waves per block.
__global__ __launch_bounds__(256) void symm_mask_kernel(
    const float* __restrict__ con, const int* __restrict__ codes,
    float* __restrict__ cm, int L) {
  const int lane = threadIdx.x & 31;
  const int wave = threadIdx.x >> 5;
  const int tps = L >> 4;                       // tiles per side
  const int tile = blockIdx.x * 8 + wave;
  if (tile >= tps * tps) return;                // wave-uniform guard (EXEC all-1)
  const int r0 = (tile / tps) << 4;
  const int c0 = (tile % tps) << 4;
  const int ln = lane & 15;
  const int hi = lane >> 4;                     // 0 or 1 (lane half)

  v8f acc1 = {};                                // M(rb,cb) in D layout
  v8f acc2 = {};                                // M(cb,rb)^T in D layout
#pragma unroll
  for (int s = 0; s < 4; ++s) {
    const int kbase = 4 * s + 2 * hi;
    // identity chunk: same lane/vgpr formula for A-layout and B-layout
    v2f id;
    id.x = (ln == (kbase + 0)) ? 1.0f : 0.0f;
    id.y = (ln == (kbase + 1)) ? 1.0f : 0.0f;
    // B operand = rows of M(rb,cb) (coalesced 64B half-rows): D1 = I x M
    v2f b;
    b.x = con[(r0 + kbase + 0) * L + c0 + ln];
    b.y = con[(r0 + kbase + 1) * L + c0 + ln];
    acc1 = __builtin_amdgcn_wmma_f32_16x16x4_f32(
        false, id, false, b, (short)0, acc1, false, false);
    // A operand = M(cb,rb) read coalesced == A-layout of its transpose:
    // D2 = (M(cb,rb)^T) x I  -> cross-lane transpose done by the matrix pipe
    v2f a;
    a.x = con[(c0 + kbase + 0) * L + r0 + ln];
    a.y = con[(c0 + kbase + 1) * L + r0 + ln];
    acc2 = __builtin_amdgcn_wmma_f32_16x16x4_f32(
        false, a, false, id, (short)0, acc2, false, false);
  }
#pragma unroll
  for (int v = 0; v < 8; ++v) {
    const int row = r0 + v + 8 * hi;            // D layout: vgpr v, lane half
    const int col = c0 + ln;
    float val = 0.5f * (acc1[v] + acc2[v]);
    int d = row - col; if (d < 0) d = -d;
    const int prod = codes[row] * codes[col];
    const bool keep =
        (d >= MIN_DIST) && (prod == 14 || prod == 15 || prod == 35);
    cm[row * L + col] = keep ? val : 0.0f;
  }
}

// ---- K3: per-row best (value desc, pair-index asc) ------------------------
__global__ __launch_bounds__(256) void rowbest_kernel(
    const float* __restrict__ cm, float* __restrict__ rbVal,
    int* __restrict__ rbCol, int L) {
  const int lane = threadIdx.x & 31;
  const int wave = threadIdx.x >> 5;
  const int r = blockIdx.x * 8 + wave;
  if (r >= L) return;
  const float* rowp = cm + (size_t)r * L;
  float bv = -__builtin_inff();
  unsigned bp = 0xFFFFFFFFu;
  int bc = -1;
  for (int c = lane; c < L; c += 32) {
    const float v = rowp[c];
    const unsigned p = (unsigned)((r < c) ? (r * L + c) : (c * L + r));
    if (v > bv || (v == bv && p < bp)) { bv = v; bp = p; bc = c; }
  }
  for (int off = 16; off > 0; off >>= 1) {
    const float v2 = __shfl_xor(bv, off, 32);
    const unsigned p2 = (unsigned)__shfl_xor((int)bp, off, 32);
    const int c2 = __shfl_xor(bc, off, 32);
    if (v2 > bv || (v2 == bv && p2 < bp)) { bv = v2; bp = p2; bc = c2; }
  }
  if (lane == 0) { rbVal[r] = bv; rbCol[r] = (bc < 0) ? r : bc; }
}

// ---- K4: greedy matching, single workgroup, lazy row-max invalidation -----
__global__ __launch_bounds__(1024) void greedy_kernel(
    const float* __restrict__ cm, const float* __restrict__ rbVal,
    const int* __restrict__ rbCol, int* __restrict__ pairI,
    int* __restrict__ pairJ, float* __restrict__ pairV,
    int* __restrict__ pnpairs, int L) {
  __shared__ float bVal[2048];
  __shared__ int bCol[2048];
  __shared__ int taken[2048];
  __shared__ float rv[1024];
  __shared__ unsigned rp[1024];
  __shared__ int ri[1024];
  __shared__ int s_cmd, s_row, s_npairs;
  const int t = threadIdx.x;
  const int gt = L / 2;

  for (int r = t; r < L; r += 1024) {
    bVal[r] = rbVal[r]; bCol[r] = rbCol[r]; taken[r] = 0;
  }
  if (t == 0) { s_npairs = 0; s_cmd = 0; }
  __syncthreads();

  int guard = 0;
  while (true) {
    // global best among free rows: (value desc, pairIdx asc)
    float v = -__builtin_inff();
    unsigned p = 0xFFFFFFFFu;
    int row = -1;
#pragma unroll
    for (int k = 0; k < 2; ++k) {
      const int r = t + k * 1024;
      if (!taken[r]) {
        const float vv = bVal[r];
        const int c = bCol[r];
        const unsigned pp = (unsigned)((r < c) ? (r * L + c) : (c * L + r));
        if (vv > v || (vv == v && pp < p)) { v = vv; p = pp; row = r; }
      }
    }
    rv[t] = v; rp[t] = p; ri[t] = row;
    __syncthreads();
    for (int off = 512; off > 0; off >>= 1) {
      if (t < off) {
        const float v2 = rv[t + off];
        const unsigned p2 = rp[t + off];
        if (v2 > rv[t] || (v2 == rv[t] && p2 < rp[t])) {
          rv[t] = v2; rp[t] = p2; ri[t] = ri[t + off];
        }
      }
      __syncthreads();
    }
    if (t == 0) {
      const float bv = rv[0];
      const int br = ri[0];
      if (br < 0 || bv <= 0.0f || s_npairs >= gt) {
        s_cmd = 2;  // done: remaining picks are zeros/negatives -> output 0
      } else {
        const int c = bCol[br];
        if (taken[c]) {
          s_cmd = 1; s_row = br;  // stale row max: recompute
        } else {
          taken[br] = 1; taken[c] = 1;
          pairI[s_npairs] = br; pairJ[s_npairs] = c; pairV[s_npairs] = bv;
          ++s_npairs;
          s_cmd = (s_npairs >= gt) ? 2 : 0;
        }
      }
    }
    __syncthreads();
    const int cmd = s_cmd;
    if (cmd == 2) break;
    if (cmd == 1) {
      const int R = s_row;
      const float* rowp = cm + (size_t)R * L;
      float vv = -__builtin_inff();
      unsigned pp = 0xFFFFFFFFu;
      int col = -1;
#pragma unroll
      for (int k = 0; k < 2; ++k) {
        const int c2 = t + k * 1024;
        if (!taken[c2]) {
          const float x = rowp[c2];
          const unsigned q = (unsigned)((R < c2) ? (R * L + c2) : (c2 * L + R));
          if (x > vv || (x == vv && q < pp)) { vv = x; pp = q; col = c2; }
        }
      }
      rv[t] = vv; rp[t] = pp; ri[t] = col;
      __syncthreads();
      for (int off = 512; off > 0; off >>= 1) {
        if (t < off) {
          const float v2 = rv[t + off];
          const unsigned p2 = rp[t + off];
          if (v2 > rv[t] || (v2 == rv[t] && p2 < rp[t])) {
            rv[t] = v2; rp[t] = p2; ri[t] = ri[t + off];
          }
        }
        __syncthreads();
      }
      if (t == 0) { bVal[R] = rv[0]; bCol[R] = (ri[0] < 0) ? R : ri[0]; }
      __syncthreads();
    }
    if (++guard > 400000) break;  // deterministic hard bound
  }
  if (t == 0) *pnpairs = s_npairs;
}

// ---- K5: zero output ------------------------------------------------------
__global__ void zero_kernel(float4* __restrict__ out, int n4) {
  const int i = blockIdx.x * blockDim.x + threadIdx.x;
  if (i < n4) out[i] = make_float4(0.0f, 0.0f, 0.0f, 0.0f);
}

// ---- K6: scatter selected pairs ------------------------------------------
__global__ void scatter_kernel(const int* __restrict__ pairI,
                               const int* __restrict__ pairJ,
                               const float* __restrict__ pairV,
                               const int* __restrict__ pnpairs,
                               float* __restrict__ out, int L) {
  const int p = threadIdx.x;
  if (p < *pnpairs) {
    const int i = pairI[p], j = pairJ[p];
    const float v = pairV[p];               // cm is exactly symmetric
    out[i * L + j] = v;
    out[j * L + i] = v;
  }
}

extern "C" void kernel_launch(void* const* d_in, const int* in_sizes, int n_in,
                              void* d_out, int out_size, void* d_ws,
                              size_t ws_size, hipStream_t stream) {
  const float* con = (const float*)d_in[0];
  const float* feat = (const float*)d_in[1];
  float* out = (float*)d_out;

  int L = 1;
  while ((long long)L * L < (long long)in_sizes[0]) L <<= 1;  // L = 2048

  char* ws = (char*)d_ws;
  int* codes   = (int*)(ws);                 // L ints
  float* rbVal = (float*)(ws + 4 * L);       // L floats
  int* rbCol   = (int*)(ws + 8 * L);         // L ints
  int* pairI   = (int*)(ws + 12 * L);        // L/2 ints
  int* pairJ   = (int*)(ws + 14 * L);        // L/2 ints
  float* pairV = (float*)(ws + 16 * L);      // L/2 floats
  int* pnp     = (int*)(ws + 18 * L);        // 1 int

  float* cm = out;  // masked symmetric matrix staged in d_out, zeroed later

  const int tiles = (L / 16) * (L / 16);
  pairs_kernel<<<(L + 255) / 256, 256, 0, stream>>>(feat, codes, L);
  symm_mask_kernel<<<(tiles + 7) / 8, 256, 0, stream>>>(con, codes, cm, L);
  rowbest_kernel<<<(L + 7) / 8, 256, 0, stream>>>(cm, rbVal, rbCol, L);
  greedy_kernel<<<1, 1024, 0, stream>>>(cm, rbVal, rbCol, pairI, pairJ, pairV,
                                        pnp, L);
  const int n4 = (L * L) / 4;
  zero_kernel<<<(n4 + 255) / 256, 256, 0, stream>>>((float4*)out, n4);
  scatter_kernel<<<1, 1024, 0, stream>>>(pairI, pairJ, pairV, pnp, out, L);
}